// LocalLossBlockLinear_90589450207575
// MI455X (gfx1250) — compile-verified
//
#include <hip/hip_runtime.h>

// Problem sizes (fixed by the reference setup)
#define SNN_B    1024
#define SNN_NIN  2048
#define SNN_NOUT 2048
#define SNN_NC   10
#define SNN_T    10

#define TDECAY  0.2f
#define TTHRESH 0.5f

typedef __attribute__((ext_vector_type(2))) float v2f;
typedef __attribute__((ext_vector_type(8))) float v8f;

#if defined(__AMDGCN__) && __has_builtin(__builtin_amdgcn_wmma_f32_16x16x4_f32)
#define USE_WMMA_F32 1
#else
#define USE_WMMA_F32 0
#endif

// CDNA5 async Global->LDS path (ASYNCcnt-tracked); guarded so either toolchain compiles.
#if defined(__AMDGCN__) && __has_builtin(__builtin_amdgcn_global_load_async_to_lds_b128) && \
    __has_builtin(__builtin_amdgcn_s_wait_asynccnt)
#define USE_ASYNC_LDS 1
// Builtin signature (from the compiler diagnostic): first arg is a global
// (address_space(1)) pointer to a 16-byte int vector; LDS side is AS(3).
typedef int vsi4 __attribute__((vector_size(4 * sizeof(int))));
typedef __attribute__((address_space(1))) vsi4 gvsi4;
typedef __attribute__((address_space(3))) vsi4 lvsi4;
#else
#define USE_ASYNC_LDS 0
#endif

// ---------------- Kernel 1: x @ W_enc^T fused with LIF time loop ----------------
// Macro tile: 16 (M) x 128 (N), K-tile 64. 256 threads = 8 wave32; each wave owns
// one 16x16 C tile (512 x V_WMMA_F32_16X16X4_F32 per tile over K=2048).
// One K-tile = A(16x64) + B(128x64) floats = 36 KB = exactly 9 B128 granules per
// thread -> uniform ASYNCcnt bookkeeping for the double-buffered async pipeline.
// Padded LDS row stride of 68 floats keeps B128 stores 16B-aligned and makes the
// 16-lane column reads bank-conflict-free (bank = 4*row + k, distinct).
#define BM  16
#define BN  128
#define BKT 64
#define LDP 68
#define GRANULES_PER_THREAD 9   // (16*64 + 128*64) floats / 4 / 256 threads

__global__ __launch_bounds__(256) void snn_encode_lif(
    const float* __restrict__ x, const float* __restrict__ Wenc,
    float* __restrict__ spikes /* [T][B][NOUT] */)
{
  __shared__ __align__(16) float As[2][BM][LDP];   //  8,704 B
  __shared__ __align__(16) float Bs[2][BN][LDP];   // 69,632 B

  const int tid  = threadIdx.x;
  const int lane = tid & 31;
  const int wave = tid >> 5;
  const int m0   = blockIdx.y * BM;
  const int n0   = blockIdx.x * BN;
  const int nloc = wave * 16;

  v8f acc = {0.f, 0.f, 0.f, 0.f, 0.f, 0.f, 0.f, 0.f};

  const int KT = SNN_NIN / BKT;  // 32 K-tiles

  // Copy one K-tile into LDS buffer `buf`. j==0 -> A region, j>=1 -> B region
  // (wave-uniform branching, 9 x 16B granules per thread).
  auto copyTile = [&](int buf, int kt) {
    const int k0 = kt * BKT;
#if !USE_ASYNC_LDS
    float4 tmp[GRANULES_PER_THREAD];
#endif
#pragma unroll
    for (int j = 0; j < GRANULES_PER_THREAD; ++j) {
      const float* gsrc;
      float* ldst;
      if (j == 0) {                       // A: 16 rows x 16 granules
        const int ar = tid >> 4, ac = (tid & 15) * 4;
        gsrc = x + (size_t)(m0 + ar) * SNN_NIN + k0 + ac;
        ldst = &As[buf][ar][ac];
      } else {                            // B: 128 rows x 16 granules
        const int gb = tid + 256 * (j - 1);
        const int br = gb >> 4, bc = (gb & 15) * 4;
        gsrc = Wenc + (size_t)(n0 + br) * SNN_NIN + k0 + bc;
        ldst = &Bs[buf][br][bc];
      }
#if USE_ASYNC_LDS
      __builtin_amdgcn_global_load_async_to_lds_b128(
          (gvsi4*)gsrc, (lvsi4*)ldst, 0, 0);
#else
      tmp[j] = *(const float4*)gsrc;
      (void)ldst;
#endif
    }
#if !USE_ASYNC_LDS
#pragma unroll
    for (int j = 0; j < GRANULES_PER_THREAD; ++j) {
      if (j == 0) {
        const int ar = tid >> 4, ac = (tid & 15) * 4;
        *(float4*)&As[buf][ar][ac] = tmp[j];
      } else {
        const int gb = tid + 256 * (j - 1);
        const int br = gb >> 4, bc = (gb & 15) * 4;
        *(float4*)&Bs[buf][br][bc] = tmp[j];
      }
    }
#endif
  };

  auto computeTile = [&](int buf) {
#if USE_WMMA_F32
    // ISA 7.12.2 layouts:
    //  A 16x4 : lane holds M = lane%16; K pair = k0 + 2*(lane/16) + {0,1}
    //  B 4x16 : lane holds N = lane%16; same K-pair selection
#pragma unroll
    for (int kk = 0; kk < BKT; kk += 4) {
      const int ko = kk + 2 * (lane >> 4);
      v2f av = *(const v2f*)&As[buf][lane & 15][ko];
      v2f bv = *(const v2f*)&Bs[buf][nloc + (lane & 15)][ko];
      acc = __builtin_amdgcn_wmma_f32_16x16x4_f32(
          false, av, false, bv, (short)0, acc, false, false);
    }
#else
    // Scalar fallback with the identical C-fragment mapping
    // (acc[i] -> M = i + 8*(lane/16), N = lane%16). Also used by the host pass.
#pragma unroll
    for (int kk = 0; kk < BKT; ++kk) {
      const float bval = Bs[buf][nloc + (lane & 15)][kk];
      const int mh = 8 * (lane >> 4);
#pragma unroll
      for (int i = 0; i < 8; ++i)
        acc[i] += As[buf][mh + i][kk] * bval;
    }
#endif
  };

  // Double-buffered pipeline. Async ops retire in order per wave: after issuing
  // the 9 ops of tile kt+1, s_wait_asynccnt(9) guarantees tile kt is fully in LDS.
  copyTile(0, 0);
  for (int kt = 0; kt < KT; ++kt) {
    if (kt + 1 < KT) {
      copyTile((kt + 1) & 1, kt + 1);
#if USE_ASYNC_LDS
      __builtin_amdgcn_s_wait_asynccnt(GRANULES_PER_THREAD);
#endif
    } else {
#if USE_ASYNC_LDS
      __builtin_amdgcn_s_wait_asynccnt(0);
#endif
    }
    __syncthreads();              // tile kt visible to all waves
    computeTile(kt & 1);
    __syncthreads();              // all waves done before buffer is overwritten
  }

  // ---- Fused LIF epilogue: h stays in registers, write spikes for all T steps.
  // C layout: acc[i] -> M = m0 + i + 8*(lane/16), N = n0 + nloc + lane%16.
  float mem[8], sp[8];
#pragma unroll
  for (int i = 0; i < 8; ++i) { mem[i] = 0.f; sp[i] = 0.f; }
  const int n  = n0 + nloc + (lane & 15);
  const int mb = m0 + 8 * (lane >> 4);
#pragma unroll
  for (int t = 0; t < SNN_T; ++t) {
#pragma unroll
    for (int i = 0; i < 8; ++i) {
      mem[i] = mem[i] * TDECAY + acc[i] - sp[i] * (TTHRESH * TDECAY);
      sp[i]  = (mem[i] > TTHRESH) ? 1.f : 0.f;
      spikes[((size_t)t * SNN_B + (mb + i)) * SNN_NOUT + n] = sp[i];
    }
  }
}

// ---------------- Kernel 2: decoder + MSE loss, one block per batch row ----------------
__global__ __launch_bounds__(256) void snn_decode_loss(
    const float* __restrict__ spikes, const float* __restrict__ Wdec,
    const float* __restrict__ y1h, float* __restrict__ loss_partial)
{
  __shared__ float wred[8][SNN_NC];
  const int b    = blockIdx.x;
  const int tid  = threadIdx.x;
  const int lane = tid & 31;
  const int wave = tid >> 5;

  float ym[SNN_NC], ys[SNN_NC];
#pragma unroll
  for (int c = 0; c < SNN_NC; ++c) { ym[c] = 0.f; ys[c] = 0.f; }
  float lossacc = 0.f;

  for (int t = 0; t < SNN_T; ++t) {
    const float* srow = spikes + ((size_t)t * SNN_B + b) * SNN_NOUT;
    float acc[SNN_NC];
#pragma unroll
    for (int c = 0; c < SNN_NC; ++c) acc[c] = 0.f;
#pragma unroll
    for (int j = 0; j < SNN_NOUT / 256; ++j) {
      const int n   = tid + 256 * j;          // coalesced; spikes are L2-resident
      const float s = srow[n];
#pragma unroll
      for (int c = 0; c < SNN_NC; ++c)
        acc[c] += s * Wdec[c * SNN_NOUT + n]; // W_dec is L2-resident (80 KB)
    }
    // wave32 reduction, then cross-wave combine in LDS (deterministic)
#pragma unroll
    for (int c = 0; c < SNN_NC; ++c) {
      float v = acc[c];
#pragma unroll
      for (int off = 16; off >= 1; off >>= 1)
        v += __shfl_xor(v, off, 32);
      acc[c] = v;
    }
    if (lane == 0) {
#pragma unroll
      for (int c = 0; c < SNN_NC; ++c) wred[wave][c] = acc[c];
    }
    __syncthreads();
    // every thread redundantly advances the tiny sequential state (no broadcast needed)
#pragma unroll
    for (int c = 0; c < SNN_NC; ++c) {
      float d = 0.f;
#pragma unroll
      for (int w = 0; w < 8; ++w) d += wred[w][c];
      ym[c] = ym[c] * TDECAY + d - ys[c] * (TTHRESH * TDECAY);
      ys[c] = (ym[c] > TTHRESH) ? 1.f : 0.f;
      const float e = ys[c] - y1h[b * SNN_NC + c];
      lossacc += e * e;
    }
    __syncthreads();
  }
  if (tid == 0)
    loss_partial[b] = lossacc * (1.0f / (float)(SNN_B * SNN_NC)); // jnp.mean
}

// ---------------- Kernel 3: deterministic final loss reduction ----------------
__global__ __launch_bounds__(256) void snn_loss_reduce(
    const float* __restrict__ part, float* __restrict__ out)
{
  __shared__ float s[256];
  const int tid = threadIdx.x;
  float v = 0.f;
  for (int i = tid; i < SNN_B; i += 256) v += part[i];
  s[tid] = v;
  __syncthreads();
  for (int off = 128; off > 0; off >>= 1) {
    if (tid < off) s[tid] += s[tid + off];
    __syncthreads();
  }
  if (tid == 0) out[0] = s[0];
}

extern "C" void kernel_launch(void* const* d_in, const int* in_sizes, int n_in,
                              void* d_out, int out_size, void* d_ws, size_t ws_size,
                              hipStream_t stream) {
  (void)in_sizes; (void)n_in; (void)out_size; (void)ws_size;
  const float* x    = (const float*)d_in[0];  // [1024, 2048]
  const float* Wenc = (const float*)d_in[1];  // [2048, 2048]
  const float* Wdec = (const float*)d_in[2];  // [10, 2048]
  const float* y1h  = (const float*)d_in[3];  // [1024, 10]
  // d_in[4] (y) and d_in[5] (time_window=10) are not needed on device.

  float* spikes = (float*)d_out;              // [10][1024][2048], then loss scalar
  float* lossp  = (float*)d_ws;               // 1024 partials (4 KB scratch)

  dim3 g1(SNN_NOUT / BN, SNN_B / BM);         // (16, 64) blocks
  snn_encode_lif<<<g1, 256, 0, stream>>>(x, Wenc, spikes);
  snn_decode_loss<<<SNN_B, 256, 0, stream>>>(spikes, Wdec, y1h, lossp);
  snn_loss_reduce<<<1, 256, 0, stream>>>(lossp,
      spikes + (size_t)SNN_T * SNN_B * SNN_NOUT);
}